// NFNDoubleQuantizer_70360154243711
// MI455X (gfx1250) — compile-verified
//
#include <hip/hip_runtime.h>
#include <stdint.h>

// NF4 double-quantize + dequantize, 4096x4096 f32.
// Memory-bound streaming op: TDM (tensor_load_to_lds / tensor_store_from_lds)
// stages whole rows through LDS with double buffering, wave32 shuffle
// reductions compute block/row statistics, VALU ladder does the codebook snap.

#define N_ROWS   4096
#define N_COLS   4096
#define R_PER_WG 4
#define TPB      256
#define EPSQ     1e-8f

// TDM pads LDS on load: 4 dwords of pad after every 16 dwords stored.
// Thread t's 16 floats live at padded float offset 20*t (16B aligned).
#define ROW_LDS_FLOATS (TPB * 20)   // 5120 floats = 20 KiB per row buffer

typedef unsigned int u32x4 __attribute__((ext_vector_type(4)));
typedef int          i32x8 __attribute__((ext_vector_type(8)));
typedef int          i32x4 __attribute__((ext_vector_type(4)));

// ---- Tensor Data Mover: issue one 1x4096-f32 row transfer -------------------
// g1w0: data_size=4B (2<<16); load adds pad_enable|pad_interval=16dw|pad_amount=4dw
//       load  -> 0x06D20000,  store -> 0x00020000
__device__ __forceinline__ void tdm_issue_row(unsigned long long gaddr,
                                              unsigned lds_off,
                                              unsigned g1w0,
                                              bool is_store) {
  u32x4 g0;
  g0.x = 1u;                                                      // count=1, user desc
  g0.y = lds_off;                                                 // lds_addr (bytes)
  g0.z = (unsigned)(gaddr & 0xFFFFFFFFull);                       // global_addr[31:0]
  g0.w = ((unsigned)((gaddr >> 32) & 0x01FFFFFFull)) | 0x80000000u; // [56:32] | type=2

  i32x8 g1;
  g1[0] = (int)g1w0;          // wg_mask=0, data_size=4B, (pad cfg for loads)
  g1[1] = (int)0x10000000;    // tensor_dim0 = 4096 (low16 in bits[31:16])
  g1[2] = (int)0x00010000;    // tensor_dim0 hi16 = 0, tensor_dim1 = 1
  g1[3] = (int)0x10000000;    // tensor_dim1 hi16 = 0, tile_dim0 = 4096
  g1[4] = 1;                  // tile_dim1 = 1, tile_dim2 = 0
  g1[5] = 4096;               // tensor_dim0_stride low32
  g1[6] = (int)0x10000000;    // stride0 hi16 = 0, tensor_dim1_stride low16 = 4096
  g1[7] = 0;                  // stride1 hi32 = 0

  i32x4 z4 = {0, 0, 0, 0};    // groups 2/3: unused dims (tile_dim2/3/4 = 0)
#if __has_include(<hip/amd_detail/amd_gfx1250_TDM.h>)
  // amdgpu-toolchain (therock-10.0 headers): 6-arg builtin form
  i32x8 z8 = {0, 0, 0, 0, 0, 0, 0, 0};
  if (is_store) __builtin_amdgcn_tensor_store_from_lds(g0, g1, z4, z4, z8, 0);
  else          __builtin_amdgcn_tensor_load_to_lds  (g0, g1, z4, z4, z8, 0);
#else
  // ROCm 7.2 clang-22: 5-arg builtin form (probe-confirmed)
  if (is_store) __builtin_amdgcn_tensor_store_from_lds(g0, g1, z4, z4, 0);
  else          __builtin_amdgcn_tensor_load_to_lds  (g0, g1, z4, z4, 0);
#endif
}

// ---- NormalFloat-4 codebook: ndtri((i+0.5)/16) / ndtri(31/32) ---------------
__device__ __forceinline__ float nf4_nearest(float x) {
  const float T[16] = {
    -1.0f,      -0.707579f, -0.542209f, -0.416823f,
    -0.311055f, -0.215946f, -0.127350f, -0.042096f,
     0.042096f,  0.127350f,  0.215946f,  0.311055f,
     0.416823f,  0.542209f,  0.707579f,  1.0f };
  float v = T[0];
#pragma unroll
  for (int i = 1; i < 16; ++i) {
    // strict '>' at the midpoint reproduces argmin's first-index tie-break
    v = (x > 0.5f * (T[i - 1] + T[i])) ? T[i] : v;
  }
  return v;
}

__device__ __forceinline__ float min4(float4 v) { return fminf(fminf(v.x, v.y), fminf(v.z, v.w)); }
__device__ __forceinline__ float max4(float4 v) { return fmaxf(fmaxf(v.x, v.y), fmaxf(v.z, v.w)); }

__device__ __forceinline__ float4 dq4(float4 a, float bmin, float inv, float hs) {
  float4 o;
  o.x = fmaf(nf4_nearest(fmaf(a.x - bmin, inv, -1.0f)) + 1.0f, hs, bmin);
  o.y = fmaf(nf4_nearest(fmaf(a.y - bmin, inv, -1.0f)) + 1.0f, hs, bmin);
  o.z = fmaf(nf4_nearest(fmaf(a.z - bmin, inv, -1.0f)) + 1.0f, hs, bmin);
  o.w = fmaf(nf4_nearest(fmaf(a.w - bmin, inv, -1.0f)) + 1.0f, hs, bmin);
  return o;
}

__global__ __launch_bounds__(TPB) void nf4_dq_kernel(const float* __restrict__ x,
                                                     float* __restrict__ out) {
  __shared__ alignas(16) float in_buf[2][ROW_LDS_FLOATS];  // 2 x 20 KiB (TDM-padded)
  __shared__ alignas(16) float out_buf[N_COLS];            // 16 KiB staging for TDM store
  __shared__ float s_scale[64];
  __shared__ float s_bmin[64];
  __shared__ float s_rowmm[2];

  const int  t      = (int)threadIdx.x;
  const bool issuer = (t < 32);                // wave 0 drives the TDM
  const int  row0   = (int)blockIdx.x * R_PER_WG;

  const unsigned in_lds0 = (unsigned)(uintptr_t)&in_buf[0][0];
  const unsigned in_lds1 = (unsigned)(uintptr_t)&in_buf[1][0];
  const unsigned out_lds = (unsigned)(uintptr_t)&out_buf[0];

  if (issuer) {
    tdm_issue_row((unsigned long long)(uintptr_t)(x + (size_t)row0 * N_COLS),
                  in_lds0, 0x06D20000u, false);
  }

  for (int r = 0; r < R_PER_WG; ++r) {
    // ---- pipeline: prefetch row r+1, wait for row r (and store r-1) --------
    if (issuer) {
      if (r + 1 < R_PER_WG) {
        tdm_issue_row((unsigned long long)(uintptr_t)(x + (size_t)(row0 + r + 1) * N_COLS),
                      ((r + 1) & 1) ? in_lds1 : in_lds0, 0x06D20000u, false);
        // In-order TENSORcnt: <=1 outstanding leaves only load(r+1) in flight,
        // so load(r) AND store(r-1) are retired (out_buf is free to rewrite).
        __builtin_amdgcn_s_wait_tensorcnt(1);
      } else {
        __builtin_amdgcn_s_wait_tensorcnt(0);
      }
    }
    __syncthreads();

    // ---- read this thread's 16 floats from the padded LDS row --------------
    const float* src = &in_buf[r & 1][t * 20];
    float4 a0 = *(const float4*)(src + 0);
    float4 a1 = *(const float4*)(src + 4);
    float4 a2 = *(const float4*)(src + 8);
    float4 a3 = *(const float4*)(src + 12);

    // ---- per-block (64 elems = 4 lanes) min/max via wave32 shuffles --------
    float mn = fminf(fminf(min4(a0), min4(a1)), fminf(min4(a2), min4(a3)));
    float mx = fmaxf(fmaxf(max4(a0), max4(a1)), fmaxf(max4(a2), max4(a3)));
    mn = fminf(mn, __shfl_xor(mn, 1, 32));
    mx = fmaxf(mx, __shfl_xor(mx, 1, 32));
    mn = fminf(mn, __shfl_xor(mn, 2, 32));
    mx = fmaxf(mx, __shfl_xor(mx, 2, 32));

    const int b = t >> 2;                         // block index 0..63
    if ((t & 3) == 0) { s_scale[b] = mx - mn; s_bmin[b] = mn; }
    __syncthreads();

    // ---- row-level scale min/max (double quantization), wave 0 only --------
    if (t < 32) {
      float sa = s_scale[t], sb = s_scale[t + 32];
      float rmn = fminf(sa, sb), rmx = fmaxf(sa, sb);
#pragma unroll
      for (int off = 16; off >= 1; off >>= 1) {
        rmn = fminf(rmn, __shfl_xor(rmn, off, 32));
        rmx = fmaxf(rmx, __shfl_xor(rmx, off, 32));
      }
      if (t == 0) { s_rowmm[0] = rmn; s_rowmm[1] = rmx; }
    }
    __syncthreads();

    // ---- dequant parameters -------------------------------------------------
    const float smin  = s_rowmm[0];
    const float smax  = s_rowmm[1];
    const float scale = s_scale[b];
    const float bmin  = s_bmin[b];
    const float sq    = rintf((scale - smin) / (smax - smin + EPSQ) * 255.0f);
    const float sdq   = smin + sq * (1.0f / 255.0f) * (smax - smin);
    const float inv   = 2.0f / (scale + EPSQ);
    const float hs    = 0.5f * sdq;

    // ---- quantize->dequantize 16 elements, stage to LDS ---------------------
    float* dst = &out_buf[t * 16];
    *(float4*)(dst + 0)  = dq4(a0, bmin, inv, hs);
    *(float4*)(dst + 4)  = dq4(a1, bmin, inv, hs);
    *(float4*)(dst + 8)  = dq4(a2, bmin, inv, hs);
    *(float4*)(dst + 12) = dq4(a3, bmin, inv, hs);
    __syncthreads();

    // ---- TDM store of the finished row (drained by next iter's wait /
    //      s_endpgm's implicit wait-idle on the last one) ---------------------
    if (issuer) {
      tdm_issue_row((unsigned long long)(uintptr_t)(out + (size_t)(row0 + r) * N_COLS),
                    out_lds, 0x00020000u, true);
    }
  }
}

extern "C" void kernel_launch(void* const* d_in, const int* in_sizes, int n_in,
                              void* d_out, int out_size, void* d_ws, size_t ws_size,
                              hipStream_t stream) {
  (void)in_sizes; (void)n_in; (void)d_ws; (void)ws_size; (void)out_size;
  const float* x  = (const float*)d_in[0];
  float*       o  = (float*)d_out;
  dim3 grid(N_ROWS / R_PER_WG);   // 1024 workgroups x 4 rows each
  dim3 block(TPB);                // 8 wave32 waves per WGP workgroup
  hipLaunchKernelGGL(nf4_dq_kernel, grid, block, 0, stream, x, o);
}